// GatNet_64991445123462
// MI455X (gfx1250) — compile-verified
//
#include <hip/hip_runtime.h>
#include <math.h>

#define HH 8            // heads
#define NGRAPHS 256

typedef float v2f __attribute__((ext_vector_type(2)));
typedef float v8f __attribute__((ext_vector_type(8)));

// ---------------------------------------------------------------------------
// Exact-fp32 WMMA GEMM: Y[N x M] = X[N x K] @ W[K x M], row-major.
// One wave computes one 16x16 tile of Y via V_WMMA_F32_16X16X4_F32.
// N must be a multiple of 16 (50000 = 3125*16), M mult of 16, K mult of 4.
// ---------------------------------------------------------------------------
__global__ __launch_bounds__(256)
void gemm_wmma_f32(const float* __restrict__ X, const float* __restrict__ W,
                   float* __restrict__ Y, int N, int K, int M) {
    int wave  = (blockIdx.x * blockDim.x + threadIdx.x) >> 5;   // uniform per wave
    int lane  = threadIdx.x & 31;
    int tilesN = M >> 4;
    int tm = wave / tilesN;
    int tn = wave % tilesN;
    if (tm * 16 >= N) return;                 // wave-uniform: EXEC stays all-ones
    int half = lane >> 4;                     // 0 = lanes 0-15, 1 = lanes 16-31
    int l15  = lane & 15;
    int row  = tm * 16 + l15;                 // A row for this lane
    int col  = tn * 16 + l15;                 // B/D column for this lane

    v8f acc = {};
    for (int k = 0; k < K; k += 4) {
        // A 16x4 layout: lanes 0-15 -> K = k+0,k+1 ; lanes 16-31 -> K = k+2,k+3
        v2f a, b;
        a.x = X[row * K + k + 2 * half + 0];
        a.y = X[row * K + k + 2 * half + 1];
        // B 4x16 layout mirrors A: one K-row striped across lanes per VGPR
        b.x = W[(k + 2 * half + 0) * M + col];
        b.y = W[(k + 2 * half + 1) * M + col];
        acc = __builtin_amdgcn_wmma_f32_16x16x4_f32(
                  false, a, false, b, (short)0, acc, false, false);
    }
    // D 16x16: VGPR r holds M=r (lanes 0-15) / M=r+8 (lanes 16-31), N=lane&15
#pragma unroll
    for (int r = 0; r < 8; ++r)
        Y[(tm * 16 + r + 8 * half) * M + col] = acc[r];
}

// ---------------------------------------------------------------------------
__global__ void fill_kernel(float* __restrict__ p, float v, int n) {
    int t = blockIdx.x * blockDim.x + threadIdx.x;
    if (t < n) p[t] = v;
}

// al_s[n,h] = sum_f h[n,h,f]*a_src[h,f] ; al_d likewise
__global__ void logits_kernel(const float* __restrict__ hbuf,
                              const float* __restrict__ a_src,
                              const float* __restrict__ a_dst,
                              float* __restrict__ al_s, float* __restrict__ al_d,
                              int N, int F) {
    int t = blockIdx.x * blockDim.x + threadIdx.x;
    if (t >= N * HH) return;
    int n = t >> 3, h = t & 7;
    const float* hp = hbuf + (size_t)n * (HH * F) + h * F;
    float ss = 0.f, sd = 0.f;
    for (int f = 0; f < F; ++f) {
        float v = hp[f];
        ss += v * a_src[h * F + f];
        sd += v * a_dst[h * F + f];
    }
    al_s[t] = ss;
    al_d[t] = sd;
}

__device__ inline void atomicMaxF(float* addr, float value) {
    // sign-split trick: signed-int max for >=0, unsigned min for <0
    if (__float_as_int(value) >= 0)
        atomicMax((int*)addr, __float_as_int(value));
    else
        atomicMin((unsigned int*)addr, __float_as_uint(value));
}

__device__ inline void edge_sd(int e, int E, const int* __restrict__ src,
                               const int* __restrict__ dst, int& s, int& d) {
    if (e < E) { s = src[e]; d = dst[e]; }
    else       { s = e - E;  d = e - E; }          // self-loops
}

__global__ void edge_max_kernel(const int* __restrict__ src, const int* __restrict__ dst,
                                int E, int N,
                                const float* __restrict__ al_s,
                                const float* __restrict__ al_d,
                                float* __restrict__ mbuf) {
    int t = blockIdx.x * blockDim.x + threadIdx.x;
    if (t >= (E + N) * HH) return;
    int h = t & 7, e = t >> 3, s, d;
    edge_sd(e, E, src, dst, s, d);
    float x = al_s[s * HH + h] + al_d[d * HH + h];
    float lr = x > 0.f ? x : 0.2f * x;             // leaky_relu(0.2)
    atomicMaxF(&mbuf[d * HH + h], lr);
}

__global__ void edge_sum_kernel(const int* __restrict__ src, const int* __restrict__ dst,
                                int E, int N,
                                const float* __restrict__ al_s,
                                const float* __restrict__ al_d,
                                const float* __restrict__ mbuf,
                                float* __restrict__ sbuf) {
    int t = blockIdx.x * blockDim.x + threadIdx.x;
    if (t >= (E + N) * HH) return;
    int h = t & 7, e = t >> 3, s, d;
    edge_sd(e, E, src, dst, s, d);
    float x = al_s[s * HH + h] + al_d[d * HH + h];
    float lr = x > 0.f ? x : 0.2f * x;
    atomicAdd(&sbuf[d * HH + h], expf(lr - mbuf[d * HH + h]));
}

__global__ void edge_agg_kernel(const int* __restrict__ src, const int* __restrict__ dst,
                                int E, int N,
                                const float* __restrict__ al_s,
                                const float* __restrict__ al_d,
                                const float* __restrict__ mbuf,
                                const float* __restrict__ sbuf,
                                const float* __restrict__ hbuf,
                                float* __restrict__ outbuf, int F) {
    int t = blockIdx.x * blockDim.x + threadIdx.x;
    if (t >= (E + N) * HH) return;
    int h = t & 7, e = t >> 3, s, d;
    edge_sd(e, E, src, dst, s, d);
    float x  = al_s[s * HH + h] + al_d[d * HH + h];
    float lr = x > 0.f ? x : 0.2f * x;
    float p  = expf(lr - mbuf[d * HH + h]);
    float alpha = p / (sbuf[d * HH + h] + 1e-16f);
    int M = HH * F;
    const float* hp = hbuf  + (size_t)s * M + h * F;
    float*       op = outbuf + (size_t)d * M + h * F;
    for (int f = 0; f < F; ++f)
        atomicAdd(&op[f], alpha * hp[f]);
}

__global__ void bias_elu_kernel(float* __restrict__ buf, const float* __restrict__ b,
                                int N, int M) {
    int t = blockIdx.x * blockDim.x + threadIdx.x;
    if (t >= N * M) return;
    float v = buf[t] + b[t % M];
    buf[t] = v > 0.f ? v : expm1f(v);              // ELU(alpha=1)
}

__global__ void pool_kernel(const float* __restrict__ feat, const int* __restrict__ batch,
                            float* __restrict__ pooled, float* __restrict__ cnt, int N) {
    int t = blockIdx.x * blockDim.x + threadIdx.x;
    if (t >= N * 128) return;
    int n = t >> 7, c = t & 127;
    int g = batch[n];
    atomicAdd(&pooled[g * 128 + c], feat[(size_t)n * 128 + c]);
    if (c == 0) atomicAdd(&cnt[g], 1.0f);
}

__global__ void mlp_kernel(const float* __restrict__ pooled, const float* __restrict__ cnt,
                           const float* __restrict__ fc1_w, const float* __restrict__ fc1_b,
                           const float* __restrict__ fc2_w, const float* __restrict__ fc2_b,
                           float* __restrict__ out) {
    int g = blockIdx.x * blockDim.x + threadIdx.x;
    if (g >= NGRAPHS) return;
    float inv = 1.0f / fmaxf(cnt[g], 1.0f);
    float acc2 = fc2_b[0];
    for (int j = 0; j < 32; ++j) {
        float acc = fc1_b[j];
        for (int i = 0; i < 128; ++i)
            acc += pooled[g * 128 + i] * inv * fc1_w[i * 32 + j];
        acc2 += fmaxf(acc, 0.0f) * fc2_w[j];
    }
    out[g] = acc2;
}

// ---------------------------------------------------------------------------
extern "C" void kernel_launch(void* const* d_in, const int* in_sizes, int n_in,
                              void* d_out, int out_size, void* d_ws, size_t ws_size,
                              hipStream_t stream) {
    const float* x      = (const float*)d_in[0];
    const int*   edge   = (const int*)  d_in[1];
    const int*   batch  = (const int*)  d_in[2];
    const float* W1     = (const float*)d_in[3];
    const float* a_s1   = (const float*)d_in[4];
    const float* a_d1   = (const float*)d_in[5];
    const float* b1     = (const float*)d_in[6];
    const float* W2     = (const float*)d_in[7];
    const float* a_s2   = (const float*)d_in[8];
    const float* a_d2   = (const float*)d_in[9];
    const float* b2     = (const float*)d_in[10];
    const float* W3     = (const float*)d_in[11];
    const float* a_s3   = (const float*)d_in[12];
    const float* a_d3   = (const float*)d_in[13];
    const float* b3     = (const float*)d_in[14];
    const float* fc1_w  = (const float*)d_in[15];
    const float* fc1_b  = (const float*)d_in[16];
    const float* fc2_w  = (const float*)d_in[17];
    const float* fc2_b  = (const float*)d_in[18];

    const int N = in_sizes[0] / 16;          // 50000
    const int E = in_sizes[1] / 2;           // 800000
    const int* src = edge;
    const int* dst = edge + E;

    // workspace carve-out (floats)
    float* ws     = (float*)d_ws;
    float* bufA   = ws;                       // [N*128] transformed features h
    float* bufB   = bufA + (size_t)N * 128;   // [N*128] layer out (ping)
    float* bufC   = bufB + (size_t)N * 128;   // [N*128] layer out (pong)
    float* al_s   = bufC + (size_t)N * 128;   // [N*8]
    float* al_d   = al_s + (size_t)N * HH;    // [N*8]
    float* mbuf   = al_d + (size_t)N * HH;    // [N*8]
    float* sbuf   = mbuf + (size_t)N * HH;    // [N*8]
    float* pooled = sbuf + (size_t)N * HH;    // [256*128]
    float* cnt    = pooled + NGRAPHS * 128;   // [256]

    auto cdiv = [](int a, int b) { return (a + b - 1) / b; };
    const int B = 256;

    auto run_layer = [&](const float* in_feat, int K, int F,
                         const float* W, const float* as, const float* ad,
                         const float* b, float* outbuf) {
        int M = HH * F;
        int waves = (N / 16) * (M / 16);
        gemm_wmma_f32<<<cdiv(waves * 32, B), B, 0, stream>>>(in_feat, W, bufA, N, K, M);
        logits_kernel<<<cdiv(N * HH, B), B, 0, stream>>>(bufA, as, ad, al_s, al_d, N, F);
        fill_kernel<<<cdiv(N * HH, B), B, 0, stream>>>(mbuf, -INFINITY, N * HH);
        fill_kernel<<<cdiv(N * HH, B), B, 0, stream>>>(sbuf, 0.0f, N * HH);
        fill_kernel<<<cdiv(N * M, B), B, 0, stream>>>(outbuf, 0.0f, N * M);
        int EH = (E + N) * HH;
        edge_max_kernel<<<cdiv(EH, B), B, 0, stream>>>(src, dst, E, N, al_s, al_d, mbuf);
        edge_sum_kernel<<<cdiv(EH, B), B, 0, stream>>>(src, dst, E, N, al_s, al_d, mbuf, sbuf);
        edge_agg_kernel<<<cdiv(EH, B), B, 0, stream>>>(src, dst, E, N, al_s, al_d,
                                                       mbuf, sbuf, bufA, outbuf, F);
        bias_elu_kernel<<<cdiv(N * M, B), B, 0, stream>>>(outbuf, b, N, M);
    };

    run_layer(x,    16,  8,  W1, a_s1, a_d1, b1, bufB);   // 16  -> 64
    run_layer(bufB, 64,  16, W2, a_s2, a_d2, b2, bufC);   // 64  -> 128
    run_layer(bufC, 128, 16, W3, a_s3, a_d3, b3, bufB);   // 128 -> 128

    fill_kernel<<<cdiv(NGRAPHS * 128, B), B, 0, stream>>>(pooled, 0.0f, NGRAPHS * 128);
    fill_kernel<<<1, B, 0, stream>>>(cnt, 0.0f, NGRAPHS);
    pool_kernel<<<cdiv(N * 128, B), B, 0, stream>>>(bufB, batch, pooled, cnt, N);
    mlp_kernel<<<1, B, 0, stream>>>(pooled, cnt, fc1_w, fc1_b, fc2_w, fc2_b, (float*)d_out);
}